// GEN_91018946936859
// MI455X (gfx1250) — compile-verified
//
#include <hip/hip_runtime.h>

#define MSG_EPS 1e-7f
#define SM_EPS  1e-16f
#define BN_EPS  1e-5f

typedef float v2f __attribute__((ext_vector_type(2)));
typedef float v8f __attribute__((ext_vector_type(8)));

static __device__ __forceinline__ v8f wmma_f32x4(v2f a, v2f b, v8f c) {
  // D = A(16x4 f32) * B(4x16 f32) + C(16x16 f32), exact f32 math
  return __builtin_amdgcn_wmma_f32_16x16x4_f32(false, a, false, b, (short)0, c,
                                               false, false);
}

// ---------------- Layer 1: node projections via WMMA (K=3 padded to 4) -----
__global__ void k_node_proj_l1(const float* __restrict__ x,
                               const float* __restrict__ wsrc, const float* __restrict__ bsrc,
                               const float* __restrict__ wdst, const float* __restrict__ bdst,
                               float* __restrict__ h, float* __restrict__ hd,
                               int N, int ntiles) {
  int wave = (int)((blockIdx.x * blockDim.x + threadIdx.x) >> 5);
  int lane = threadIdx.x & 31;
  if (wave >= ntiles) return;               // uniform per wave
  int base = wave * 16;
  bool full = (base + 16 <= N);             // uniform per wave
  int m = lane & 15;
  bool hi = lane >= 16;
  int col = m;
  int row = base + m; if (row >= N) row = N - 1;

  v2f a;                                     // A: lane<16 -> K0,K1 ; lane>=16 -> K2,K3(=0)
  a.x = hi ? x[row * 3 + 2] : x[row * 3 + 0];
  a.y = hi ? 0.0f          : x[row * 3 + 1];

  // ---- src projection ----
  v2f b;
  b.x = hi ? wsrc[2 * 16 + col] : wsrc[0 * 16 + col];
  b.y = hi ? 0.0f               : wsrc[1 * 16 + col];
  v8f c;
  float bias = bsrc[col];
  #pragma unroll
  for (int r = 0; r < 8; ++r) c[r] = bias;
  c = wmma_f32x4(a, b, c);
  int rbase = base + (hi ? 8 : 0);
  if (full) {
    #pragma unroll
    for (int r = 0; r < 8; ++r) h[(rbase + r) * 16 + col] = c[r];
  } else {
    #pragma unroll
    for (int r = 0; r < 8; ++r) if (rbase + r < N) h[(rbase + r) * 16 + col] = c[r];
  }

  // ---- dst projection ----
  b.x = hi ? wdst[2 * 16 + col] : wdst[0 * 16 + col];
  b.y = hi ? 0.0f               : wdst[1 * 16 + col];
  bias = bdst[col];
  #pragma unroll
  for (int r = 0; r < 8; ++r) c[r] = bias;
  c = wmma_f32x4(a, b, c);
  if (full) {
    #pragma unroll
    for (int r = 0; r < 8; ++r) hd[(rbase + r) * 16 + col] = c[r];
  } else {
    #pragma unroll
    for (int r = 0; r < 8; ++r) if (rbase + r < N) hd[(rbase + r) * 16 + col] = c[r];
  }
}

// ---------------- Layer 1: edge pass 1 (segment max via int atomicMax) -----
__global__ void k_edge_pass1_l1(const int* __restrict__ ei, const float* __restrict__ ea,
                                const float* __restrict__ wedge, const float* __restrict__ bedge,
                                const float* __restrict__ h, int* __restrict__ mx, int E) {
  int e = blockIdx.x * 256 + threadIdx.x;
  if (e >= E) return;
  int src = ei[e], dst = ei[E + e];
  const float2* ea2 = (const float2*)(ea + (size_t)e * 6);   // 8B aligned (e*24 bytes)
  float2 p0 = ea2[0], p1 = ea2[1], p2 = ea2[2];
  float at[6] = {p0.x, p0.y, p1.x, p1.y, p2.x, p2.y};
  const float4* hs4 = (const float4*)(h + (size_t)src * 16); // 64B aligned
  float4 h0 = hs4[0], h1 = hs4[1], h2 = hs4[2], h3 = hs4[3];
  float hs[16] = {h0.x, h0.y, h0.z, h0.w, h1.x, h1.y, h1.z, h1.w,
                  h2.x, h2.y, h2.z, h2.w, h3.x, h3.y, h3.z, h3.w};
  #pragma unroll
  for (int c = 0; c < 16; ++c) {
    float v = bedge[c];
    #pragma unroll
    for (int k = 0; k < 6; ++k) v += at[k] * wedge[k * 16 + c];
    float msg = fmaxf(hs[c] + v, 0.0f) + MSG_EPS;   // > 0 always
    atomicMax(&mx[(size_t)dst * 16 + c], __float_as_int(msg));
  }
}

// ------- Layer 1: edge pass 2 (denom += ex ; numer += msg*ex) --------------
__global__ void k_edge_pass2_l1(const int* __restrict__ ei, const float* __restrict__ ea,
                                const float* __restrict__ wedge, const float* __restrict__ bedge,
                                const float* __restrict__ h, const float* __restrict__ mx,
                                float* __restrict__ den, float* __restrict__ num, int E) {
  int e = blockIdx.x * 256 + threadIdx.x;
  if (e >= E) return;
  int src = ei[e], dst = ei[E + e];
  const float2* ea2 = (const float2*)(ea + (size_t)e * 6);
  float2 p0 = ea2[0], p1 = ea2[1], p2 = ea2[2];
  float at[6] = {p0.x, p0.y, p1.x, p1.y, p2.x, p2.y};
  const float4* hs4 = (const float4*)(h + (size_t)src * 16);
  float4 h0 = hs4[0], h1 = hs4[1], h2 = hs4[2], h3 = hs4[3];
  float hs[16] = {h0.x, h0.y, h0.z, h0.w, h1.x, h1.y, h1.z, h1.w,
                  h2.x, h2.y, h2.z, h2.w, h3.x, h3.y, h3.z, h3.w};
  const float4* mx4 = (const float4*)(mx + (size_t)dst * 16);
  float4 m0 = mx4[0], m1 = mx4[1], m2 = mx4[2], m3 = mx4[3];
  float mxv[16] = {m0.x, m0.y, m0.z, m0.w, m1.x, m1.y, m1.z, m1.w,
                   m2.x, m2.y, m2.z, m2.w, m3.x, m3.y, m3.z, m3.w};
  #pragma unroll
  for (int c = 0; c < 16; ++c) {
    float v = bedge[c];
    #pragma unroll
    for (int k = 0; k < 6; ++k) v += at[k] * wedge[k * 16 + c];
    float msg = fmaxf(hs[c] + v, 0.0f) + MSG_EPS;
    float ex = expf(msg - mxv[c]);
    atomicAdd(&den[(size_t)dst * 16 + c], ex);
    atomicAdd(&num[(size_t)dst * 16 + c], msg * ex);
  }
}

// --- Layer 1 MLP stage 1 (fused combine): z = (agg+hd)@w1 + b1, BN partials -
__global__ void k_mlp1_z(const float* __restrict__ num, const float* __restrict__ den,
                         const float* __restrict__ hd,
                         const float* __restrict__ w1, const float* __restrict__ b1,
                         float* __restrict__ z,
                         float* __restrict__ bnsum, float* __restrict__ bnsq,
                         int N, int ntiles) {
  __shared__ float s_sum[32], s_sq[32];
  if (threadIdx.x < 32) { s_sum[threadIdx.x] = 0.f; s_sq[threadIdx.x] = 0.f; }
  __syncthreads();

  int wave = (int)((blockIdx.x * blockDim.x + threadIdx.x) >> 5);
  int lane = threadIdx.x & 31;
  bool active = wave < ntiles;
  if (active) {
    int base = wave * 16;
    bool full = (base + 16 <= N);
    int m = lane & 15;
    bool hi = lane >= 16;
    int col = m;
    int row = base + m; if (row >= N) row = N - 1;

    // A operand: out = num/(den+eps) + hd, computed inline, loaded once
    float av[8];
    #pragma unroll
    for (int kk = 0; kk < 4; ++kk) {
      int k0 = kk * 4 + (hi ? 2 : 0);
      size_t i0 = (size_t)row * 16 + k0;
      av[2 * kk]     = num[i0]     / (den[i0]     + SM_EPS) + hd[i0];
      av[2 * kk + 1] = num[i0 + 1] / (den[i0 + 1] + SM_EPS) + hd[i0 + 1];
    }

    #pragma unroll
    for (int ct = 0; ct < 2; ++ct) {
      v8f acc;
      float bias = b1[ct * 16 + col];
      #pragma unroll
      for (int r = 0; r < 8; ++r) acc[r] = bias;
      #pragma unroll
      for (int kk = 0; kk < 4; ++kk) {
        int k0 = kk * 4 + (hi ? 2 : 0);
        v2f a; a.x = av[2 * kk];                     a.y = av[2 * kk + 1];
        v2f b; b.x = w1[k0 * 32 + ct * 16 + col];    b.y = w1[(k0 + 1) * 32 + ct * 16 + col];
        acc = wmma_f32x4(a, b, acc);
      }
      float s = 0.f, q = 0.f;
      int rbase = base + (hi ? 8 : 0);
      if (full) {
        #pragma unroll
        for (int r = 0; r < 8; ++r) {
          z[(rbase + r) * 32 + ct * 16 + col] = acc[r];
          s += acc[r]; q += acc[r] * acc[r];
        }
      } else {
        #pragma unroll
        for (int r = 0; r < 8; ++r) {
          if (rbase + r < N) {
            z[(rbase + r) * 32 + ct * 16 + col] = acc[r];
            s += acc[r]; q += acc[r] * acc[r];
          }
        }
      }
      s += __shfl_xor(s, 16, 32);                    // lane L and L+16 share a channel
      q += __shfl_xor(q, 16, 32);
      if (!hi) { atomicAdd(&s_sum[ct * 16 + col], s); atomicAdd(&s_sq[ct * 16 + col], q); }
    }
  }
  __syncthreads();
  if (threadIdx.x < 32) {
    atomicAdd(&bnsum[threadIdx.x], s_sum[threadIdx.x]);
    atomicAdd(&bnsq[threadIdx.x],  s_sq[threadIdx.x]);
  }
}

// ---------------- BN finalize --------------------------------------------
__global__ void k_bn_final(const float* __restrict__ sum, const float* __restrict__ sq,
                           const float* __restrict__ g, const float* __restrict__ beta,
                           float* __restrict__ scale, float* __restrict__ shift,
                           int C, float invN) {
  int t = threadIdx.x;
  if (t < C) {
    float mean = sum[t] * invN;
    float var  = sq[t] * invN - mean * mean;
    float sc   = g[t] * rsqrtf(var + BN_EPS);
    scale[t] = sc;
    shift[t] = beta[t] - mean * sc;
  }
}

// ------ Layer 1 MLP stage 2: relu(BN(z)) @ w2 + b2, then inter-layer relu --
__global__ void k_mlp1_out(const float* __restrict__ z, const float* __restrict__ scale,
                           const float* __restrict__ shift, const float* __restrict__ w2,
                           const float* __restrict__ b2, float* __restrict__ hrelu,
                           int N, int ntiles) {
  __shared__ float s_sc[32], s_sh[32];
  if (threadIdx.x < 32) { s_sc[threadIdx.x] = scale[threadIdx.x]; s_sh[threadIdx.x] = shift[threadIdx.x]; }
  __syncthreads();

  int wave = (int)((blockIdx.x * blockDim.x + threadIdx.x) >> 5);
  int lane = threadIdx.x & 31;
  if (wave >= ntiles) return;
  int base = wave * 16;
  bool full = (base + 16 <= N);
  int m = lane & 15;
  bool hi = lane >= 16;
  int col = m;
  int row = base + m; if (row >= N) row = N - 1;

  v8f acc;
  float bias = b2[col];
  #pragma unroll
  for (int r = 0; r < 8; ++r) acc[r] = bias;
  #pragma unroll
  for (int kk = 0; kk < 8; ++kk) {
    int k0 = kk * 4 + (hi ? 2 : 0);
    v2f a;
    a.x = fmaxf(z[row * 32 + k0] * s_sc[k0] + s_sh[k0], 0.f);
    a.y = fmaxf(z[row * 32 + k0 + 1] * s_sc[k0 + 1] + s_sh[k0 + 1], 0.f);
    v2f b; b.x = w2[k0 * 16 + col]; b.y = w2[(k0 + 1) * 16 + col];
    acc = wmma_f32x4(a, b, acc);
  }
  int rbase = base + (hi ? 8 : 0);
  if (full) {
    #pragma unroll
    for (int r = 0; r < 8; ++r) hrelu[(rbase + r) * 16 + col] = fmaxf(acc[r], 0.f);
  } else {
    #pragma unroll
    for (int r = 0; r < 8; ++r)
      if (rbase + r < N) hrelu[(rbase + r) * 16 + col] = fmaxf(acc[r], 0.f);
  }
}

// ---------------------------- Layer 2 (C = 1) ------------------------------
__global__ void k_l2_node_proj(const float* __restrict__ hin,
                               const float* __restrict__ wsrc, const float* __restrict__ bsrc,
                               const float* __restrict__ wdst, const float* __restrict__ bdst,
                               float* __restrict__ h2, float* __restrict__ hd2, int N) {
  int i = blockIdx.x * 256 + threadIdx.x;
  if (i >= N) return;
  const float4* p4 = (const float4*)(hin + (size_t)i * 16);
  float4 r0 = p4[0], r1 = p4[1], r2 = p4[2], r3 = p4[3];
  float v[16] = {r0.x, r0.y, r0.z, r0.w, r1.x, r1.y, r1.z, r1.w,
                 r2.x, r2.y, r2.z, r2.w, r3.x, r3.y, r3.z, r3.w};
  float a = bsrc[0], d = bdst[0];
  #pragma unroll
  for (int k = 0; k < 16; ++k) { a += v[k] * wsrc[k]; d += v[k] * wdst[k]; }
  h2[i] = a; hd2[i] = d;
}

__global__ void k_edge_pass1_l2(const int* __restrict__ ei, const float* __restrict__ ea,
                                const float* __restrict__ wedge, const float* __restrict__ bedge,
                                const float* __restrict__ h2, int* __restrict__ mx2, int E) {
  int e = blockIdx.x * 256 + threadIdx.x;
  if (e >= E) return;
  int src = ei[e], dst = ei[E + e];
  const float2* ea2 = (const float2*)(ea + (size_t)e * 6);
  float2 p0 = ea2[0], p1 = ea2[1], p2 = ea2[2];
  float v = bedge[0] + p0.x * wedge[0] + p0.y * wedge[1] + p1.x * wedge[2]
                     + p1.y * wedge[3] + p2.x * wedge[4] + p2.y * wedge[5];
  float msg = fmaxf(h2[src] + v, 0.f) + MSG_EPS;
  atomicMax(&mx2[dst], __float_as_int(msg));
}

__global__ void k_edge_pass2_l2(const int* __restrict__ ei, const float* __restrict__ ea,
                                const float* __restrict__ wedge, const float* __restrict__ bedge,
                                const float* __restrict__ h2, const float* __restrict__ mx2,
                                float* __restrict__ den2, float* __restrict__ num2, int E) {
  int e = blockIdx.x * 256 + threadIdx.x;
  if (e >= E) return;
  int src = ei[e], dst = ei[E + e];
  const float2* ea2 = (const float2*)(ea + (size_t)e * 6);
  float2 p0 = ea2[0], p1 = ea2[1], p2 = ea2[2];
  float v = bedge[0] + p0.x * wedge[0] + p0.y * wedge[1] + p1.x * wedge[2]
                     + p1.y * wedge[3] + p2.x * wedge[4] + p2.y * wedge[5];
  float msg = fmaxf(h2[src] + v, 0.f) + MSG_EPS;
  float ex = expf(msg - mx2[dst]);
  atomicAdd(&den2[dst], ex);
  atomicAdd(&num2[dst], msg * ex);
}

__global__ void k_l2_combine(const float* __restrict__ num2, const float* __restrict__ den2,
                             const float* __restrict__ hd2, const float* __restrict__ w1,
                             const float* __restrict__ b1, float* __restrict__ z2,
                             float* __restrict__ bnacc, int N) {
  __shared__ float s[4];
  if (threadIdx.x < 4) s[threadIdx.x] = 0.f;
  __syncthreads();
  int i = blockIdx.x * 256 + threadIdx.x;
  float z0 = 0.f, z1 = 0.f, q0 = 0.f, q1 = 0.f;
  if (i < N) {
    float outv = num2[i] / (den2[i] + SM_EPS) + hd2[i];
    z0 = outv * w1[0] + b1[0];
    z1 = outv * w1[1] + b1[1];
    z2[2 * i] = z0; z2[2 * i + 1] = z1;
    q0 = z0 * z0; q1 = z1 * z1;
  }
  #pragma unroll
  for (int off = 16; off > 0; off >>= 1) {
    z0 += __shfl_xor(z0, off, 32); z1 += __shfl_xor(z1, off, 32);
    q0 += __shfl_xor(q0, off, 32); q1 += __shfl_xor(q1, off, 32);
  }
  if ((threadIdx.x & 31) == 0) {
    atomicAdd(&s[0], z0); atomicAdd(&s[1], z1);
    atomicAdd(&s[2], q0); atomicAdd(&s[3], q1);
  }
  __syncthreads();
  if (threadIdx.x < 4) atomicAdd(&bnacc[threadIdx.x], s[threadIdx.x]);
}

__global__ void k_l2_bn_final(const float* __restrict__ acc, const float* __restrict__ g,
                              const float* __restrict__ beta, float* __restrict__ ss, float invN) {
  int t = threadIdx.x;
  if (t < 2) {
    float mean = acc[t] * invN;
    float var  = acc[2 + t] * invN - mean * mean;
    float sc   = g[t] * rsqrtf(var + BN_EPS);
    ss[t] = sc;
    ss[2 + t] = beta[t] - mean * sc;
  }
}

__global__ void k_l2_out(const float* __restrict__ z2, const float* __restrict__ ss,
                         const float* __restrict__ w2, const float* __restrict__ b2,
                         float* __restrict__ out, int N) {
  int i = blockIdx.x * 256 + threadIdx.x;
  if (i >= N) return;
  float a0 = fmaxf(z2[2 * i] * ss[0] + ss[2], 0.f);
  float a1 = fmaxf(z2[2 * i + 1] * ss[1] + ss[3], 0.f);
  float y = a0 * w2[0] + a1 * w2[1] + b2[0];
  out[i] = 1.f / (1.f + expf(-y));
}

// ---------------------------------------------------------------------------
extern "C" void kernel_launch(void* const* d_in, const int* in_sizes, int n_in,
                              void* d_out, int out_size, void* d_ws, size_t ws_size,
                              hipStream_t stream) {
  (void)n_in; (void)out_size; (void)ws_size;
  const int N = in_sizes[0] / 3;
  const int E = in_sizes[2] / 6;

  const float* x        = (const float*)d_in[0];
  const int*   ei       = (const int*)  d_in[1];
  const float* ea       = (const float*)d_in[2];
  const float* c1_wsrc  = (const float*)d_in[3];
  const float* c1_bsrc  = (const float*)d_in[4];
  const float* c1_wedge = (const float*)d_in[5];
  const float* c1_bedge = (const float*)d_in[6];
  const float* c1_wdst  = (const float*)d_in[7];
  const float* c1_bdst  = (const float*)d_in[8];
  const float* c1_w1    = (const float*)d_in[9];
  const float* c1_b1    = (const float*)d_in[10];
  const float* c1_w2    = (const float*)d_in[11];
  const float* c1_b2    = (const float*)d_in[12];
  const float* c1_g     = (const float*)d_in[13];
  const float* c1_beta  = (const float*)d_in[14];
  const float* c2_wsrc  = (const float*)d_in[15];
  const float* c2_bsrc  = (const float*)d_in[16];
  const float* c2_wedge = (const float*)d_in[17];
  const float* c2_bedge = (const float*)d_in[18];
  const float* c2_wdst  = (const float*)d_in[19];
  const float* c2_bdst  = (const float*)d_in[20];
  const float* c2_w1    = (const float*)d_in[21];
  const float* c2_b1    = (const float*)d_in[22];
  const float* c2_w2    = (const float*)d_in[23];
  const float* c2_b2    = (const float*)d_in[24];
  const float* c2_g     = (const float*)d_in[25];
  const float* c2_beta  = (const float*)d_in[26];
  float* out = (float*)d_out;

  float* F = (float*)d_ws;
  size_t n = (size_t)N;
  float* h1   = F;               // [N,16]  (reused later as hrelu)
  float* hd1  = F + 16 * n;      // [N,16]
  float* mx1  = F + 32 * n;      // [N,16]
  float* den1 = F + 48 * n;      // [N,16]
  float* num1 = F + 64 * n;      // [N,16]
  float* z1   = F + 80 * n;      // [N,32]
  float* h2   = F + 112 * n;     // [N]
  float* hd2  = F + 113 * n;     // [N]
  float* mx2  = F + 114 * n;     // [N]
  float* den2 = F + 115 * n;     // [N]
  float* num2 = F + 116 * n;     // [N]
  float* z2   = F + 117 * n;     // [N,2]
  float* st   = F + 119 * n;     // stats: 136 floats
  float* bn1sum = st;       float* bn1sq = st + 32;
  float* bn1sc  = st + 64;  float* bn1sh = st + 96;
  float* bn2acc = st + 128; float* bn2ss = st + 132;

  hipMemsetAsync(mx1, 0, sizeof(float) * 48 * n, stream);   // mx1, den1, num1
  hipMemsetAsync(mx2, 0, sizeof(float) * 3 * n, stream);    // mx2, den2, num2
  hipMemsetAsync(st,  0, sizeof(float) * 136, stream);

  int ntiles = (N + 15) / 16;
  int tb = (ntiles + 7) / 8;            // 8 waves (16-node tiles) per block
  int eg = (E + 255) / 256;
  int ng = (N + 255) / 256;

  // ----- Layer 1 -----
  k_node_proj_l1<<<tb, 256, 0, stream>>>(x, c1_wsrc, c1_bsrc, c1_wdst, c1_bdst, h1, hd1, N, ntiles);
  k_edge_pass1_l1<<<eg, 256, 0, stream>>>(ei, ea, c1_wedge, c1_bedge, h1, (int*)mx1, E);
  k_edge_pass2_l1<<<eg, 256, 0, stream>>>(ei, ea, c1_wedge, c1_bedge, h1, mx1, den1, num1, E);
  k_mlp1_z<<<tb, 256, 0, stream>>>(num1, den1, hd1, c1_w1, c1_b1, z1, bn1sum, bn1sq, N, ntiles);
  k_bn_final<<<1, 32, 0, stream>>>(bn1sum, bn1sq, c1_g, c1_beta, bn1sc, bn1sh, 32, 1.0f / (float)N);
  k_mlp1_out<<<tb, 256, 0, stream>>>(z1, bn1sc, bn1sh, c1_w2, c1_b2, /*hrelu=*/h1, N, ntiles);

  // ----- Layer 2 -----
  k_l2_node_proj<<<ng, 256, 0, stream>>>(h1, c2_wsrc, c2_bsrc, c2_wdst, c2_bdst, h2, hd2, N);
  k_edge_pass1_l2<<<eg, 256, 0, stream>>>(ei, ea, c2_wedge, c2_bedge, h2, (int*)mx2, E);
  k_edge_pass2_l2<<<eg, 256, 0, stream>>>(ei, ea, c2_wedge, c2_bedge, h2, mx2, den2, num2, E);
  k_l2_combine<<<ng, 256, 0, stream>>>(num2, den2, hd2, c2_w1, c2_b1, z2, bn2acc, N);
  k_l2_bn_final<<<1, 32, 0, stream>>>(bn2acc, c2_g, c2_beta, bn2ss, 1.0f / (float)N);
  k_l2_out<<<ng, 256, 0, stream>>>(z2, bn2ss, c2_w2, c2_b2, out, N);
}